// TemporalDifference_27625229647901
// MI455X (gfx1250) — compile-verified
//
#include <hip/hip_runtime.h>

// ---------------------------------------------------------------------------
// TD objective:  td = mean_b  sum_t ( e[t,b] - G[t,b] )^2
//   with G_t = r_t + GAMMA * G_{t+1},  G_T = 0   (reverse scan over T)
// T = 8192, B = 4096, float32.  Memory-bound: 256 MB streamed once (~11 us
// floor at 23.3 TB/s).  Chunked linear-scan decomposition for parallelism;
// 4-deep async global->LDS ring (CDNA5 ASYNCcnt path) keeps ~3KB/wave in
// flight (~24 MB across 8192 waves) to cover HBM latency at full bandwidth.
// ---------------------------------------------------------------------------

#define GAMMA          0.99f
#define T_DIM          8192
#define B_DIM          4096
#define CHUNK          128                  // L: rows per chunk
#define NCHUNK         (T_DIM / CHUNK)      // 64
#define ROWS_PER_STAGE 4
#define NSTAGE         (CHUNK / ROWS_PER_STAGE) // 32
#define COLS_PER_WAVE  32
#define WAVES_PER_BLK  8
#define THREADS        (WAVES_PER_BLK * 32) // 256
#define BTILES         (B_DIM / COLS_PER_WAVE) // 128
#define PIPE           4                    // async pipeline depth (ring bufs)

#if defined(__HIP_DEVICE_COMPILE__) && defined(__gfx1250__) && \
    __has_builtin(__builtin_amdgcn_global_load_async_to_lds_b128)
#define HAVE_ASYNC 1
#else
#define HAVE_ASYNC 0
#endif

#if HAVE_ASYNC
// builtin signature (from clang diagnostic): (v4i AS1*, v4i AS3*, imm, imm)
typedef int v4i __attribute__((vector_size(16)));
typedef __attribute__((address_space(1))) v4i* as1_v4i_p;
typedef __attribute__((address_space(3))) v4i* as3_v4i_p;
#endif

__device__ __forceinline__ void copy16_g2lds(const float* g, float* l) {
#if HAVE_ASYNC
  // async DMA: global (AS1) -> LDS (AS3), 16 bytes per lane, ASYNCcnt-tracked
  __builtin_amdgcn_global_load_async_to_lds_b128(
      (as1_v4i_p)g, (as3_v4i_p)l, /*imm offset*/ 0, /*cpol*/ 0);
#else
  float4 v = *(const float4*)g;
  *(float4*)l = v;
#endif
}

template <int N>
__device__ __forceinline__ void wait_async() {
#if HAVE_ASYNC
#if __has_builtin(__builtin_amdgcn_s_wait_asynccnt)
  __builtin_amdgcn_s_wait_asynccnt(N);
#else
  asm volatile("s_wait_asynccnt %0" ::"n"(N) : "memory");
#endif
#endif
}

__device__ __forceinline__ void stage_issue(const float* __restrict__ rew,
                                            const float* __restrict__ exr,
                                            float* ldsR, float* ldsE,
                                            int tBase, int b0, int lr, int lc) {
  const size_t go = (size_t)(tBase + lr) * B_DIM + (size_t)(b0 + lc);
  copy16_g2lds(rew + go, ldsR + (lr * COLS_PER_WAVE + lc));
  copy16_g2lds(exr + go, ldsE + (lr * COLS_PER_WAVE + lc));
}

// ---------------------------------------------------------------------------
// Pass 1: per (chunk c, column b) compute
//   A = sum d^2,  B = sum d*gamma^{end-t},  R = local return at chunk start
// where local_t = zero-carry in-chunk reverse scan, d_t = e_t - local_t.
// One wave = 32 consecutive columns of one chunk.  PIPE-deep async LDS ring,
// wave-private buffers => no barriers; in-order ASYNCcnt retires oldest stage.
// ---------------------------------------------------------------------------
__global__ void __launch_bounds__(THREADS)
td_pass1(const float* __restrict__ rewards, const float* __restrict__ exr,
         float* __restrict__ outA, float* __restrict__ outB,
         float* __restrict__ outR) {
  __shared__ float lds[WAVES_PER_BLK][PIPE][2][ROWS_PER_STAGE * COLS_PER_WAVE];

  const int lane = threadIdx.x & 31;
  const int wIn  = threadIdx.x >> 5;
  const int wgid = blockIdx.x * WAVES_PER_BLK + wIn;
  const int c    = wgid / BTILES;                // chunk index  [0,64)
  const int b0   = (wgid % BTILES) * COLS_PER_WAVE;
  const int tEnd = (c + 1) * CHUNK;

  const int lr = lane >> 3;        // row within stage   (0..3)
  const int lc = (lane & 7) << 2;  // col start (x4 floats, b128 per lane)

  // prologue: prefetch stages 0..PIPE-2 (stage s covers 4 rows ending at
  // tEnd - 4*s, processed in descending-t order)
#pragma unroll
  for (int k = 0; k < PIPE - 1; ++k) {
    stage_issue(rewards, exr, lds[wIn][k][0], lds[wIn][k][1],
                tEnd - ROWS_PER_STAGE * (k + 1), b0, lr, lc);
  }

  float gl = 0.f, p = GAMMA, accA = 0.f, accB = 0.f;

  for (int s = 0; s < NSTAGE; ++s) {
    const int cur = s & (PIPE - 1);
    if (s + (PIPE - 1) < NSTAGE) {
      stage_issue(rewards, exr, lds[wIn][(s + PIPE - 1) & (PIPE - 1)][0],
                  lds[wIn][(s + PIPE - 1) & (PIPE - 1)][1],
                  tEnd - ROWS_PER_STAGE * (s + PIPE), b0, lr, lc);
      wait_async<2 * (PIPE - 1)>();  // oldest stage (s) retired, 3 in flight
    } else if (s + 2 < NSTAGE) {
      wait_async<4>();
    } else if (s + 1 < NSTAGE) {
      wait_async<2>();
    } else {
      wait_async<0>();
    }
#pragma unroll
    for (int i = ROWS_PER_STAGE - 1; i >= 0; --i) {  // t descending
      float r = lds[wIn][cur][0][i * COLS_PER_WAVE + lane];
      float e = lds[wIn][cur][1][i * COLS_PER_WAVE + lane];
      gl = fmaf(GAMMA, gl, r);     // local_t = r_t + g*local_{t+1}
      float d = e - gl;
      accA = fmaf(d, d, accA);
      accB = fmaf(d, p, accB);
      p *= GAMMA;                  // p = gamma^{end-t}
    }
  }

  const size_t o = (size_t)c * B_DIM + (size_t)(b0 + lane);
  outA[o] = accA;
  outB[o] = accB;
  outR[o] = gl;  // local return at chunk start
}

// ---------------------------------------------------------------------------
// Pass 2: per column, reverse scan over 64 chunk carries:
//   contribution = A - 2*C*B + C^2*P ;  C' = R + gamma^L * C
// then deterministic block tree-reduction over 256 columns.
// ---------------------------------------------------------------------------
__global__ void __launch_bounds__(256)
td_pass2(const float* __restrict__ A, const float* __restrict__ B,
         const float* __restrict__ R, float* __restrict__ partials,
         float P, float gL) {
  __shared__ float red[256];
  const int b = blockIdx.x * 256 + threadIdx.x;
  float C = 0.f, acc = 0.f;
  for (int c = NCHUNK - 1; c >= 0; --c) {
    const size_t idx = (size_t)c * B_DIM + (size_t)b;
    float a = A[idx], bb = B[idx], r = R[idx];
    acc += a + C * (P * C - 2.f * bb);
    C = fmaf(gL, C, r);
  }
  red[threadIdx.x] = acc;
  __syncthreads();
  for (int off = 128; off > 0; off >>= 1) {
    if (threadIdx.x < off) red[threadIdx.x] += red[threadIdx.x + off];
    __syncthreads();
  }
  if (threadIdx.x == 0) partials[blockIdx.x] = red[0];
}

__global__ void td_pass3(const float* __restrict__ partials,
                         float* __restrict__ out, int n, float invB) {
  if (blockIdx.x == 0 && threadIdx.x == 0) {
    float s = 0.f;
    for (int i = 0; i < n; ++i) s += partials[i];
    out[0] = s * invB;
  }
}

// ---------------------------------------------------------------------------
extern "C" void kernel_launch(void* const* d_in, const int* in_sizes, int n_in,
                              void* d_out, int out_size, void* d_ws,
                              size_t ws_size, hipStream_t stream) {
  (void)in_sizes; (void)n_in; (void)out_size; (void)ws_size;
  // setup_inputs order: exp_reward first, rewards second
  const float* exr     = (const float*)d_in[0];
  const float* rewards = (const float*)d_in[1];
  float* out = (float*)d_out;

  const size_t NB = (size_t)NCHUNK * B_DIM;
  float* wsA = (float*)d_ws;
  float* wsB = wsA + NB;
  float* wsR = wsB + NB;
  float* wsP = wsR + NB;  // 16 block partials

  // exact chunk constants in double: gamma^L and P = sum_{k=1..L} gamma^{2k}
  const double gd = (double)GAMMA;
  double gL = 1.0;
  for (int i = 0; i < CHUNK; ++i) gL *= gd;
  const double g2 = gd * gd;
  double P = 0.0, tp = 1.0;
  for (int k = 0; k < CHUNK; ++k) { tp *= g2; P += tp; }

  const int nWaves  = NCHUNK * BTILES;          // 8192
  const int nBlocks = nWaves / WAVES_PER_BLK;   // 1024
  td_pass1<<<dim3(nBlocks), dim3(THREADS), 0, stream>>>(rewards, exr, wsA, wsB,
                                                        wsR);
  td_pass2<<<dim3(B_DIM / 256), dim3(256), 0, stream>>>(
      wsA, wsB, wsR, wsP, (float)P, (float)gL);
  td_pass3<<<dim3(1), dim3(32), 0, stream>>>(wsP, out, B_DIM / 256,
                                             1.0f / (float)B_DIM);
}